// PFNet_37134287241357
// MI455X (gfx1250) — compile-verified
//
#include <hip/hip_runtime.h>

// ---------------------------------------------------------------------------
// PFNet forward for MI455X (gfx1250, wave32).
//   GAT edge phase: memory/atomic bound -> coalesced gathers + f32 atomics.
//   MLP heads:      dense fp32 GEMMs    -> V_WMMA_F32_16X16X4_F32 tiles.
// ---------------------------------------------------------------------------

typedef __attribute__((ext_vector_type(2))) float v2f;
typedef __attribute__((ext_vector_type(8))) float v8f;

__device__ __forceinline__ float lrelu(float v, float sl) { return v > 0.f ? v : v * sl; }

// monotone float max/min via integer atomics (valid incl. -inf init)
__device__ __forceinline__ void atomicMaxF(float* a, float v) {
  if (v >= 0.f) atomicMax((int*)a, __float_as_int(v));
  else          atomicMin((unsigned int*)a, __float_as_uint(v));
}

// ---------------------------------------------------------------------------
// K1: per-node projection h = x @ W (K=3), plus attention dots a_src/a_dst.
// One node per 128-thread block; h lives only in LDS (recomputed later).
// ---------------------------------------------------------------------------
__global__ __launch_bounds__(128) void k_node_proj(
    const float* __restrict__ x, const float* __restrict__ W,
    const float* __restrict__ att_src, const float* __restrict__ att_dst,
    float* __restrict__ a_src, float* __restrict__ a_dst, int n)
{
  __shared__ float hs[128];
  int node = blockIdx.x;
  if (node >= n) return;
  int c = threadIdx.x;
  float x0 = x[node * 3 + 0], x1 = x[node * 3 + 1], x2 = x[node * 3 + 2];
  float hv = fmaf(x0, W[c], fmaf(x1, W[128 + c], x2 * W[256 + c]));
  hs[c] = hv;
  __syncthreads();
  if (c < 8) {
    int head = c & 3;
    const float* att = (c < 4) ? att_src : att_dst;
    float acc = 0.f;
#pragma unroll
    for (int d = 0; d < 32; ++d) acc = fmaf(hs[head * 32 + d], att[head * 32 + d], acc);
    if (c < 4) a_src[node * 4 + head] = acc;
    else       a_dst[node * 4 + head] = acc;
  }
}

// ---------------------------------------------------------------------------
// K2: init m = -inf, s = 0, acc = 0
// ---------------------------------------------------------------------------
__global__ void k_init(float* __restrict__ m, float* __restrict__ s,
                       float* __restrict__ acc, int n)
{
  int i = blockIdx.x * blockDim.x + threadIdx.x;
  if (i < n * 4) { m[i] = -__builtin_inff(); s[i] = 0.f; }
  if (i < n * 128) acc[i] = 0.f;
}

// ---------------------------------------------------------------------------
// K3: segment max of leaky_relu(a_src[src]+a_dst[dst], 0.2) per (dst, head)
// ---------------------------------------------------------------------------
__global__ __launch_bounds__(256) void k_edge_max(
    const int* __restrict__ ei, int E, int n,
    const float* __restrict__ a_src, const float* __restrict__ a_dst,
    float* __restrict__ m)
{
  long long t = (long long)blockIdx.x * blockDim.x + threadIdx.x;
  long long total = (long long)(E + n) * 4;
  if (t >= total) return;
  int head = (int)(t & 3);
  int e    = (int)(t >> 2);
  int src, dst;
  if (e < E) { src = ei[e]; dst = ei[E + e]; } else { src = dst = e - E; }
  float logit = lrelu(a_src[src * 4 + head] + a_dst[dst * 4 + head], 0.2f);
  atomicMaxF(&m[dst * 4 + head], logit);
}

// ---------------------------------------------------------------------------
// K4: segment sum of exp(logit - m[dst]) per (dst, head)
// ---------------------------------------------------------------------------
__global__ __launch_bounds__(256) void k_edge_sum(
    const int* __restrict__ ei, int E, int n,
    const float* __restrict__ a_src, const float* __restrict__ a_dst,
    const float* __restrict__ m, float* __restrict__ s)
{
  long long t = (long long)blockIdx.x * blockDim.x + threadIdx.x;
  long long total = (long long)(E + n) * 4;
  if (t >= total) return;
  int head = (int)(t & 3);
  int e    = (int)(t >> 2);
  int src, dst;
  if (e < E) { src = ei[e]; dst = ei[E + e]; } else { src = dst = e - E; }
  float logit = lrelu(a_src[src * 4 + head] + a_dst[dst * 4 + head], 0.2f);
  float ex = expf(logit - m[dst * 4 + head]);
  atomicAdd(&s[dst * 4 + head], ex);
}

// ---------------------------------------------------------------------------
// K5: message scatter. One wave per (edge, head); lane d handles channel d.
// h[src] is recomputed from x[src] (12B broadcast) and the L0-hot 3x128
// weight matrix instead of a 128B gather per edge-head.
// ---------------------------------------------------------------------------
__global__ __launch_bounds__(256) void k_edge_msg(
    const int* __restrict__ ei, int E, int n,
    const float* __restrict__ x, const float* __restrict__ W,
    const float* __restrict__ a_src, const float* __restrict__ a_dst,
    const float* __restrict__ m, const float* __restrict__ s,
    float* __restrict__ acc)
{
  int wave = blockIdx.x * (blockDim.x >> 5) + (threadIdx.x >> 5);
  int lane = threadIdx.x & 31;
  int total = (E + n) * 4;
  if (wave >= total) return;
  int head = wave & 3;
  int e    = wave >> 2;
  int src, dst;
  if (e < E) { src = ei[e]; dst = ei[E + e]; } else { src = dst = e - E; }

  float logit = lrelu(a_src[src * 4 + head] + a_dst[dst * 4 + head], 0.2f);
  float alpha = expf(logit - m[dst * 4 + head]) / s[dst * 4 + head];

  int c = head * 32 + lane;
  float x0 = x[src * 3 + 0], x1 = x[src * 3 + 1], x2 = x[src * 3 + 2];
  float hv = fmaf(x0, W[c], fmaf(x1, W[128 + c], x2 * W[256 + c]));
  atomicAdd(&acc[dst * 128 + c], hv * alpha);
}

// ---------------------------------------------------------------------------
// K6: conv[n][d] = leaky_relu(mean_heads(acc) + bias, 0.01)
// ---------------------------------------------------------------------------
__global__ __launch_bounds__(256) void k_conv(
    const float* __restrict__ acc, const float* __restrict__ bias,
    float* __restrict__ conv, int n)
{
  int i = blockIdx.x * blockDim.x + threadIdx.x;
  if (i >= n * 32) return;
  int node = i >> 5, d = i & 31;
  const float* a = acc + (size_t)node * 128;
  float v = (a[d] + a[32 + d] + a[64 + d] + a[96 + d]) * 0.25f + bias[d];
  conv[i] = lrelu(v, 0.01f);
}

// ---------------------------------------------------------------------------
// K7: two 5-layer MLPs on up = [x | conv] using V_WMMA_F32_16X16X4_F32.
// One 16-node tile per wave; activations ping-pong through per-wave LDS.
// ---------------------------------------------------------------------------
struct MlpParams { const float* w[10]; const float* b[10]; };

#define UP_STRIDE 36           // 35 (=3+32) padded to multiple of 4
#define WPB 8                  // waves per block

__device__ __forceinline__ v8f wmma_ktile(
    const float* __restrict__ actL, int astride,
    const float* __restrict__ W, int Ktrue, int Ntrue, int Ksteps,
    int ncol, int lane)
{
  v8f c = {0.f, 0.f, 0.f, 0.f, 0.f, 0.f, 0.f, 0.f};
  int row  = lane & 15;
  int kofs = (lane >= 16) ? 2 : 0;       // A/B: lanes 16-31 carry K+2,K+3
  bool colok = (ncol < Ntrue);
  for (int st = 0; st < Ksteps; ++st) {
    int k = 4 * st + kofs;
    v2f a, b;
    a.x = actL[row * astride + k];
    a.y = actL[row * astride + k + 1];
    b.x = (colok && k     < Ktrue) ? W[k * Ntrue + ncol]       : 0.f;
    b.y = (colok && k + 1 < Ktrue) ? W[(k + 1) * Ntrue + ncol] : 0.f;
    c = __builtin_amdgcn_wmma_f32_16x16x4_f32(
        /*neg_a=*/false, a, /*neg_b=*/false, b,
        /*c_mod=*/(short)0, c, /*reuse_a=*/false, /*reuse_b=*/false);
  }
  return c;
}

__device__ __forceinline__ void hidden_layer(
    const float* __restrict__ actL, int astride, int Ktrue,
    const float* __restrict__ W, const float* __restrict__ bias,
    float* __restrict__ dst, int lane)
{
  int Ksteps = (Ktrue + 3) >> 2;
#pragma unroll
  for (int half = 0; half < 2; ++half) {
    int ncol = half * 16 + (lane & 15);
    v8f c = wmma_ktile(actL, astride, W, Ktrue, 32, Ksteps, ncol, lane);
    float bv = bias[ncol];
    int rbase = (lane >= 16) ? 8 : 0;
#pragma unroll
    for (int r = 0; r < 8; ++r)
      dst[(rbase + r) * 32 + ncol] = lrelu(c[r] + bv, 0.01f);
  }
}

__device__ __forceinline__ void final_layer(
    const float* __restrict__ actL,
    const float* __restrict__ W, const float* __restrict__ bias,
    float* __restrict__ out, int base, int Nout, int lane, bool valid)
{
  int ncol = lane & 15;
  v8f c = wmma_ktile(actL, 32, W, 32, Nout, 8, ncol, lane);
  if (valid && ncol < Nout) {
    float bv = bias[ncol];
    int rbase = (lane >= 16) ? 8 : 0;
#pragma unroll
    for (int r = 0; r < 8; ++r)
      out[(size_t)(base + rbase + r) * Nout + ncol] = c[r] + bv;
  }
}

__global__ __launch_bounds__(256) void k_mlp(
    const float* __restrict__ x, const float* __restrict__ conv, MlpParams p,
    float* __restrict__ out_id, float* __restrict__ out_p, int n)
{
  __shared__ float lds[WPB * (16 * UP_STRIDE + 16 * 32 * 2)];   // 50 KB
  int wave = threadIdx.x >> 5;
  int lane = threadIdx.x & 31;
  int tile = blockIdx.x * WPB + wave;
  int base = tile * 16;
  bool valid = base < n;                  // N % 16 == 0 -> tiles never split

  float* up   = lds + wave * (16 * UP_STRIDE + 1024);
  float* ping = up + 16 * UP_STRIDE;
  float* pong = ping + 512;

  // stage up = [x(3) | conv(32) | pad(1)=0]
  for (int i = lane; i < 16 * UP_STRIDE; i += 32) {
    int r = i / UP_STRIDE, cidx = i % UP_STRIDE;
    int node = base + r;
    float v = 0.f;
    if (node < n) {
      if (cidx < 3)       v = x[node * 3 + cidx];
      else if (cidx < 35) v = conv[node * 32 + (cidx - 3)];
    }
    up[i] = v;
  }
  __syncthreads();

  // MLP 1 -> cand_id [N,8]
  hidden_layer(up,   UP_STRIDE, 35, p.w[0], p.b[0], ping, lane); __syncthreads();
  hidden_layer(ping, 32,        32, p.w[1], p.b[1], pong, lane); __syncthreads();
  hidden_layer(pong, 32,        32, p.w[2], p.b[2], ping, lane); __syncthreads();
  hidden_layer(ping, 32,        32, p.w[3], p.b[3], pong, lane); __syncthreads();
  final_layer(pong, p.w[4], p.b[4], out_id, base, 8, lane, valid);
  __syncthreads();

  // MLP 2 -> cand_p [N,3]  (reuses untouched `up`)
  hidden_layer(up,   UP_STRIDE, 35, p.w[5], p.b[5], ping, lane); __syncthreads();
  hidden_layer(ping, 32,        32, p.w[6], p.b[6], pong, lane); __syncthreads();
  hidden_layer(pong, 32,        32, p.w[7], p.b[7], ping, lane); __syncthreads();
  hidden_layer(ping, 32,        32, p.w[8], p.b[8], pong, lane); __syncthreads();
  final_layer(pong, p.w[9], p.b[9], out_p, base, 3, lane, valid);
}

// ---------------------------------------------------------------------------
extern "C" void kernel_launch(void* const* d_in, const int* in_sizes, int n_in,
                              void* d_out, int out_size, void* d_ws, size_t ws_size,
                              hipStream_t stream)
{
  const float* x       = (const float*)d_in[0];
  const int*   ei      = (const int*)d_in[1];     // [2,E] int32
  // d_in[2] edge_attr, d_in[3] batch: unused by the reference forward
  const float* gat_w   = (const float*)d_in[4];
  const float* att_src = (const float*)d_in[5];
  const float* att_dst = (const float*)d_in[6];
  const float* gat_b   = (const float*)d_in[7];

  MlpParams p;
  for (int i = 0; i < 5; ++i) {
    p.w[i]     = (const float*)d_in[8 + 2 * i];
    p.b[i]     = (const float*)d_in[9 + 2 * i];
    p.w[5 + i] = (const float*)d_in[18 + 2 * i];
    p.b[5 + i] = (const float*)d_in[19 + 2 * i];
  }

  int n = in_sizes[0] / 3;
  int E = in_sizes[1] / 2;

  // workspace layout (floats): a_src | a_dst | m | s | acc(128) | conv(32)
  float* ws    = (float*)d_ws;
  float* a_src = ws;
  float* a_dst = a_src + (size_t)n * 4;
  float* m     = a_dst + (size_t)n * 4;
  float* s     = m     + (size_t)n * 4;
  float* acc   = s     + (size_t)n * 4;
  float* conv  = acc   + (size_t)n * 128;

  float* out_id = (float*)d_out;
  float* out_p  = out_id + (size_t)n * 8;

  k_node_proj<<<n, 128, 0, stream>>>(x, gat_w, att_src, att_dst, a_src, a_dst, n);

  k_init<<<(n * 128 + 255) / 256, 256, 0, stream>>>(m, s, acc, n);

  long long etot4 = (long long)(E + n) * 4;
  int eblocks = (int)((etot4 + 255) / 256);
  k_edge_max<<<eblocks, 256, 0, stream>>>(ei, E, n, a_src, a_dst, m);
  k_edge_sum<<<eblocks, 256, 0, stream>>>(ei, E, n, a_src, a_dst, m, s);

  int mblocks = (int)((etot4 + WPB - 1) / WPB);
  k_edge_msg<<<mblocks, 256, 0, stream>>>(ei, E, n, x, gat_w, a_src, a_dst, m, s, acc);

  k_conv<<<(n * 32 + 255) / 256, 256, 0, stream>>>(acc, gat_b, conv, n);

  int tiles = (n + 15) / 16;
  k_mlp<<<(tiles + WPB - 1) / WPB, 256, 0, stream>>>(x, conv, p, out_id, out_p, n);
}